// QKRotationWrapper_1675037245588
// MI455X (gfx1250) — compile-verified
//
#include <hip/hip_runtime.h>
#include <math.h>

typedef __attribute__((ext_vector_type(2))) float v2f;
typedef __attribute__((ext_vector_type(8))) float v8f;

#define BSZ 2
#define HN  32
#define SN  2048
#define DN  128
#define LDS_STRIDE 132               // 128 + 4 dword pad: conflict-free strided reads
#define HADSCL 0.08838834764831845f  // 1/sqrt(128)

// One wave computes Y[16x128] = X[16x128] * (H128/sqrt(128)) using the
// Kronecker factorization H128 = H8 (x) H16:
//   stage 1: eight K=16 GEMMs (4 x V_WMMA_F32_16X16X4_F32 each) against
//            H16-sign B fragments (only 4 distinct fragments, precomputed)
//   stage 2: radix-2 FWHT butterfly across the 8 accumulator tiles (VALU)
// X tile lives in LDS (row stride LDS_STRIDE floats). On return z[n] holds
// the 16x16 output tile for columns [16n, 16n+16).
//
// A 16x4 f32 layout: lanes 0-15 hold {K0,K1} of row M=lane; lanes 16-31 hold
// {K2,K3}. C/D 16x16 f32: VGPR v, lanes 0-15 -> (M=v, N=lane); 16-31 -> M=v+8.
__device__ __forceinline__ void wave_fwht16(const float* lds, int lane, v8f z[8]) {
    const int laneLo = lane & 15;
    const int laneHi = lane >> 4;

    // 4 B fragments of the 16-point Hadamard, scale folded in (one-time VALU)
    v2f bfrag[4];
    #pragma unroll
    for (int kc = 0; kc < 4; ++kc) {
        const int klo0 = (kc << 2) | (laneHi << 1);
        bfrag[kc].x = (__popc(klo0 & laneLo) & 1) ? -HADSCL : HADSCL;
        bfrag[kc].y = (__popc((klo0 + 1) & laneLo) & 1) ? -HADSCL : HADSCL;
    }

    // Stage 1: Z[khi] = X[:, 16*khi .. 16*khi+15] * H16 * (1/sqrt(128))
    #pragma unroll
    for (int khi = 0; khi < 8; ++khi) {
        #pragma unroll
        for (int kc = 0; kc < 4; ++kc) {
            const int col = (khi << 4) + (kc << 2) + (laneHi << 1);
            v2f a = *(const v2f*)(lds + laneLo * LDS_STRIDE + col);  // ds_load_b64
            // 8 args: (neg_a, A, neg_b, B, c_mod, C, reuse_a, reuse_b)
            z[khi] = __builtin_amdgcn_wmma_f32_16x16x4_f32(
                false, a, false, bfrag[kc], (short)0, z[khi], false, false);
        }
    }

    // Stage 2: 8-point FWHT across tiles; z[n] = sum_khi (-1)^popc(khi&n) Z[khi]
    #pragma unroll
    for (int st = 1; st < 8; st <<= 1) {
        #pragma unroll
        for (int i = 0; i < 8; ++i) {
            if (i & st) continue;
            const v8f a = z[i], b = z[i ^ st];
            z[i]      = a + b;
            z[i ^ st] = a - b;
        }
    }
}

// ---- q path: pure Hadamard transform, 64 contiguous rows per block ----
__global__ void __launch_bounds__(128) had_q_kernel(const float* __restrict__ qin,
                                                    float* __restrict__ qout,
                                                    int nblocks) {
    __shared__ float tile[64 * LDS_STRIDE];
    const int t = threadIdx.x;
    const long rowbase = (long)blockIdx.x * 64;

    // Cooperative coalesced load: 64 rows x 128 f32 = 2048 float4
    const float4* gin = (const float4*)(qin + rowbase * DN);
    #pragma unroll
    for (int i = 0; i < 16; ++i) {
        const int c4 = i * 128 + t;          // float4 index within 64x128 block
        const int r = c4 >> 5, col = c4 & 31;
        *(float4*)&tile[r * LDS_STRIDE + col * 4] = gin[c4];
    }
    // Prefetch next block's tile while we compute (global_prefetch_b8)
    if (blockIdx.x + 1 < nblocks) {
        __builtin_prefetch((const char*)(gin + 2048) + t * 256, 0, 0);
    }
    __syncthreads();

    const int wave = t >> 5, lane = t & 31;
    const int laneLo = lane & 15, laneHi = lane >> 4;

    v8f z[8] = {};
    wave_fwht16(&tile[wave * 16 * LDS_STRIDE], lane, z);

    float* gout = qout + (rowbase + wave * 16) * DN;
    #pragma unroll
    for (int n = 0; n < 8; ++n) {
        #pragma unroll
        for (int v = 0; v < 8; ++v) {
            const int m = v + laneHi * 8;
            gout[m * DN + n * 16 + laneLo] = z[n][v];
        }
    }
}

// ---- k path: Hadamard + per-token (32 heads x 128) 4-bit fake-quant ----
__global__ void __launch_bounds__(64) had_k_kernel(const float* __restrict__ kin,
                                                   float* __restrict__ kout) {
    __shared__ float tile[32 * LDS_STRIDE];
    __shared__ float wmax[2];
    const int t = threadIdx.x;
    const int token = blockIdx.x;            // 0 .. B*S-1
    const int b = token >> 11;               // S = 2048
    const int s = token & (SN - 1);

    // Gather this token's 32 head-rows (1 MB apart) with coalesced float4 loads
    #pragma unroll
    for (int i = 0; i < 16; ++i) {
        const int c4 = i * 64 + t;           // 1024 float4 total
        const int h = c4 >> 5, col = c4 & 31;
        const float4* src =
            (const float4*)(kin + ((((long)b * HN + h) * SN + s) * DN)) + col;
        *(float4*)&tile[h * LDS_STRIDE + col * 4] = *src;
        // Prefetch the next token's row chunk (same b, s+1: contiguous 512B ahead)
        if (s + 1 < SN) __builtin_prefetch((const char*)(src) + DN * 4, 0, 0);
    }
    __syncthreads();

    const int wave = t >> 5, lane = t & 31;
    const int laneLo = lane & 15, laneHi = lane >> 4;

    v8f z[8] = {};
    wave_fwht16(&tile[wave * 16 * LDS_STRIDE], lane, z);

    // Per-token max|y| across all 32 rows (both waves)
    float m = 0.0f;
    #pragma unroll
    for (int n = 0; n < 8; ++n)
        #pragma unroll
        for (int v = 0; v < 8; ++v)
            m = fmaxf(m, fabsf(z[n][v]));
    #pragma unroll
    for (int off = 16; off >= 1; off >>= 1)
        m = fmaxf(m, __shfl_xor(m, off, 32));
    if (lane == 0) wmax[wave] = m;
    __syncthreads();

    const float xmax  = fmaxf(wmax[0], wmax[1]) * 0.95f;             // K_CLIP
    const float scale = (xmax == 0.0f) ? 1.0f : xmax * (1.0f / 7.0f); // MAXQ = 7
    const float inv   = 1.0f / scale;

    // Quantize (round-half-even = v_rndne), clamp [-8, 7], dequantize, store
    #pragma unroll
    for (int n = 0; n < 8; ++n) {
        #pragma unroll
        for (int v = 0; v < 8; ++v) {
            const int h = wave * 16 + v + laneHi * 8;
            const float y  = z[n][v];
            const float qv = fminf(7.0f, fmaxf(-8.0f, rintf(y * inv)));
            kout[(((long)b * HN + h) * SN + s) * DN + n * 16 + laneLo] = qv * scale;
        }
    }
}

extern "C" void kernel_launch(void* const* d_in, const int* in_sizes, int n_in,
                              void* d_out, int out_size, void* d_ws, size_t ws_size,
                              hipStream_t stream) {
    const float* q = (const float*)d_in[0];
    const float* k = (const float*)d_in[1];
    float* out_q = (float*)d_out;
    float* out_k = out_q + (size_t)BSZ * HN * SN * DN;   // tuple order (q, k)

    const int qrows   = BSZ * HN * SN;                   // 131072 rows of 128
    const int qblocks = qrows / 64;                      // 2048
    had_q_kernel<<<qblocks, 128, 0, stream>>>(q, out_q, qblocks);
    had_k_kernel<<<BSZ * SN, 64, 0, stream>>>(k, out_k);
}